// IncrementalClassRectificationLoss_32186484917023
// MI455X (gfx1250) — compile-verified
//
#include <hip/hip_runtime.h>
#include <math.h>

#define BATCH   8192
#define NCLS    5000
#define KSEL    20
#define NCG     313          // ceil(5000/16)
#define CPAD    (NCG * 16)   // 5008
#define POSCAP  4096

typedef __attribute__((ext_vector_type(2))) float v2f;
typedef __attribute__((ext_vector_type(8))) float v8f;

struct Ws {
    double bce_sum;       // 0
    float  dp;            // 8
    float  dn;            // 12
    int    n_sel;         // 16
    int    pad[3];
    float  counts[CPAD];
    int    sel[CPAD];
};

__global__ void initws_k(Ws* ws) {
    int i = blockIdx.x * blockDim.x + threadIdx.x;
    if (i < CPAD) { ws->counts[i] = 0.0f; ws->sel[i] = 0; }
    if (i == 0) { ws->bce_sum = 0.0; ws->dp = 0.0f; ws->dn = 0.0f; ws->n_sel = 0; }
}

// log1p(exp(-a)) for a >= 0, using raw v_exp_f32 / v_log_f32 (no range fix-up
// needed: exp2 arg <= 0, log arg in [1,2]).
__device__ __forceinline__ float softplus_neg_abs(float a) {
    const float e = __builtin_amdgcn_exp2f(a * -1.44269504088896341f); // exp(-a)
    return __builtin_amdgcn_logf(1.0f + e) * 0.69314718055994531f;     // ln(1+e)
}

// Pass 1: fused BCE sum + per-class counts via V_WMMA_F32_16X16X4_F32 (A = ones).
// grid = (NCG column-groups of 16, 16 row-slabs of 512), block = 256 (8 waves).
// Each wave: 64 rows x 16 cols; 16 WMMA steps of K=4. Branchless inner loop:
// tail columns are clamped to a valid address and masked by multiply, so EXEC
// stays all-ones and loads pipeline without saveexec churn.
__global__ __launch_bounds__(256) void pass1_k(const float* __restrict__ logits,
                                               const float* __restrict__ target,
                                               Ws* __restrict__ ws) {
    const int cg   = blockIdx.x;
    const int slab = blockIdx.y;
    const int tid  = threadIdx.x;
    const int lane = tid & 31;
    const int wave = tid >> 5;
    const int col  = cg * 16 + (lane & 15);
    const bool colOK = (col < NCLS);
    const int colc = colOK ? col : (NCLS - 1);     // clamped (always valid address)
    const float msk = colOK ? 1.0f : 0.0f;
    const int kofs = (lane >> 4) * 2;              // lanes 0-15 -> K=0,1 ; 16-31 -> K=2,3
    const int rowBase = slab * 512 + wave * 64;

    const float* lp = logits + (long)(rowBase + kofs) * NCLS + colc;
    const float* tp = target + (long)(rowBase + kofs) * NCLS + colc;

    v2f a; a[0] = 1.0f; a[1] = 1.0f;               // all-ones A (16x4)
    v8f acc = {};
    float bce = 0.0f;

    for (int it = 0; it < 16; ++it) {
        const long ofs = (long)it * (4 * NCLS);
        const float t0 = tp[ofs];
        const float t1 = tp[ofs + NCLS];
        const float l0 = lp[ofs];
        const float l1 = lp[ofs + NCLS];

        // bce(l,t) = max(l,0) - l*t + log1p(exp(-|l|))
        bce += msk * (fmaxf(l0, 0.0f) - l0 * t0 + softplus_neg_abs(fabsf(l0))
                    + fmaxf(l1, 0.0f) - l1 * t1 + softplus_neg_abs(fabsf(l1)));

        v2f b; b[0] = t0 * msk; b[1] = t1 * msk;   // B 4x16, lane carries column N
        acc = __builtin_amdgcn_wmma_f32_16x16x4_f32(false, a, false, b,
                                                    (short)0, acc, false, false);
    }

    // D[m,n] identical for all m (A==1). Row M=0 lives in acc[0], lanes 0-15, N=lane.
    if ((lane < 16) && colOK) atomicAdd(&ws->counts[col], acc[0]);

    // wave + block reduce of BCE, one f64 atomic per block
    for (int off = 16; off > 0; off >>= 1) bce += __shfl_xor(bce, off, 32);
    __shared__ float wsum[8];
    if (lane == 0) wsum[wave] = bce;
    __syncthreads();
    if (tid == 0) {
        float s = 0.0f;
        for (int w = 0; w < 8; ++w) s += wsum[w];
        atomicAdd(&ws->bce_sum, (double)s);
    }
}

// Pass 2: stable-argsort-equivalent minority selection.
__global__ void pass2_k(Ws* __restrict__ ws) {
    const int j = blockIdx.x * blockDim.x + threadIdx.x;
    if (j >= NCLS) return;
    const float cj = ws->counts[j];
    float s_less = 0.0f;
    int ties = 0;
    for (int i = 0; i < NCLS; ++i) {
        const float ci = ws->counts[i];
        if (ci < cj) s_less += ci;
        else if (ci == cj && i < j) ++ties;
    }
    const float cum = s_less + cj * (float)(ties + 1);
    if (cum <= 0.5f * (float)BATCH && cj > 1.0f) {
        const int p = atomicAdd(&ws->n_sel, 1);
        ws->sel[p] = j;
    }
}

// Pass 3: per selected class, CRL partial sums dp/dn.
__global__ __launch_bounds__(256) void pass3_k(const float* __restrict__ logits,
                                               const float* __restrict__ target,
                                               Ws* __restrict__ ws) {
    __shared__ float pos[POSCAP];
    __shared__ int   ranks[POSCAP];
    __shared__ float cand[256 * KSEL];
    __shared__ float rv[256];
    __shared__ int   ri[256];
    __shared__ float xs_small[KSEL + 1];
    __shared__ float negsel[KSEL];
    __shared__ int   pos_cnt;

    const int tid = threadIdx.x;
    const float FINF = __builtin_inff();
    const int nsel = ws->n_sel;

    for (int si = blockIdx.x; si < nsel; si += gridDim.x) {
        const int j = ws->sel[si];
        if (tid == 0) pos_cnt = 0;
        for (int k = tid; k < 256 * KSEL; k += 256) cand[k] = FINF;
        __syncthreads();

        // scan column j: gather positive sigmoids, per-thread 20 smallest negatives
        for (int r = tid; r < BATCH; r += 256) {
            const float l = logits[(long)r * NCLS + j];
            const float t = target[(long)r * NCLS + j];
            const float p = 1.0f / (1.0f + expf(-l));
            if (t == 1.0f) {
                const int ix = atomicAdd(&pos_cnt, 1);
                if (ix < POSCAP) pos[ix] = p;
            } else {
                float* my = &cand[tid * KSEL];
                if (p < my[KSEL - 1]) {
                    int q = KSEL - 1;
                    while (q > 0 && my[q - 1] > p) { my[q] = my[q - 1]; --q; }
                    my[q] = p;
                }
            }
        }
        __syncthreads();

        int c_p = pos_cnt;
        if (c_p > POSCAP) c_p = POSCAP;
        if (c_p < 1) { __syncthreads(); continue; }

        int n_n = BATCH - c_p; if (n_n > KSEL) n_n = KSEL;
        int np_a = c_p - 1;    if (np_a > KSEL) np_a = KSEL;

        // extract n_n smallest negatives (n_n rounds of block argmin)
        for (int t = 0; t < n_n; ++t) {
            float mv = FINF; int mi = -1;
            for (int k = tid; k < 256 * KSEL; k += 256) {
                const float v = cand[k];
                if (v < mv) { mv = v; mi = k; }
            }
            rv[tid] = mv; ri[tid] = mi;
            __syncthreads();
            for (int s2 = 128; s2 > 0; s2 >>= 1) {
                if (tid < s2 && rv[tid + s2] < rv[tid]) {
                    rv[tid] = rv[tid + s2]; ri[tid] = ri[tid + s2];
                }
                __syncthreads();
            }
            if (tid == 0) { negsel[t] = rv[0]; if (ri[0] >= 0) cand[ri[0]] = FINF; }
            __syncthreads();
        }

        // stable ranks by counting; scatter the (np_a+1) smallest into xs_small
        for (int i = tid; i < c_p; i += 256) {
            const float vi = pos[i];
            int rk = 0;
            for (int k = 0; k < c_p; ++k) {
                const float vk = pos[k];
                if (vk < vi || (vk == vi && k < i)) ++rk;
            }
            ranks[i] = rk;
            if (rk <= np_a) xs_small[rk] = vi;
        }
        __syncthreads();

        float lpv = 0.0f, lnv = 0.0f;
        for (int i = tid; i < c_p; i += 256) {
            const float vi = pos[i];
            const int rk = ranks[i];
            const int limit = (rk <= np_a) ? (np_a + 1) : np_a;
            for (int r = 0; r < limit; ++r)
                if (r != rk) lpv += fabsf(vi - xs_small[r]);
            for (int r = 0; r < n_n; ++r)
                lnv += fabsf(vi - negsel[r]);
        }

        rv[tid] = lpv; __syncthreads();
        for (int s2 = 128; s2 > 0; s2 >>= 1) { if (tid < s2) rv[tid] += rv[tid + s2]; __syncthreads(); }
        const float pos_sum = rv[0];
        __syncthreads();
        rv[tid] = lnv; __syncthreads();
        for (int s2 = 128; s2 > 0; s2 >>= 1) { if (tid < s2) rv[tid] += rv[tid + s2]; __syncthreads(); }
        if (tid == 0) {
            atomicAdd(&ws->dp, (float)n_n * pos_sum);
            if (n_n > 0) atomicAdd(&ws->dn, (float)np_a * rv[0]);
        }
        __syncthreads();
    }
}

__global__ void finalize_k(const Ws* __restrict__ ws, float* __restrict__ out) {
    const float bce = (float)(ws->bce_sum / (double)((long)BATCH * NCLS));
    float loss;
    if (ws->n_sel > 0) {
        float crl = ws->dp - ws->dn + 0.5f;
        crl = crl > 0.0f ? crl : 0.0f;
        loss = 0.5f * crl + 0.5f * bce;
    } else {
        loss = bce;
    }
    out[0] = loss;
}

extern "C" void kernel_launch(void* const* d_in, const int* in_sizes, int n_in,
                              void* d_out, int out_size, void* d_ws, size_t ws_size,
                              hipStream_t stream) {
    const float* logits = (const float*)d_in[0];
    const float* target = (const float*)d_in[1];
    Ws* ws = (Ws*)d_ws;
    float* out = (float*)d_out;

    initws_k<<<(CPAD + 255) / 256, 256, 0, stream>>>(ws);

    dim3 g1(NCG, 16);
    pass1_k<<<g1, 256, 0, stream>>>(logits, target, ws);

    pass2_k<<<(NCLS + 255) / 256, 256, 0, stream>>>(ws);

    pass3_k<<<64, 256, 0, stream>>>(logits, target, ws);

    finalize_k<<<1, 1, 0, stream>>>(ws, out);
}